// Geo_GCN_16741782520368
// MI455X (gfx1250) — compile-verified
//
#include <hip/hip_runtime.h>

#define D 64

typedef __attribute__((ext_vector_type(2))) float v2f;
typedef __attribute__((ext_vector_type(8))) float v8f;

__global__ void zero_f32(float* __restrict__ p, int n) {
    int i = blockIdx.x * blockDim.x + threadIdx.x;
    if (i < n) p[i] = 0.0f;
}

// deg[col[e]] += 1
__global__ void degree_kernel(const int* __restrict__ col, float* __restrict__ deg, int E) {
    int e = blockIdx.x * blockDim.x + threadIdx.x;
    if (e < E) atomicAdd(&deg[col[e]], 1.0f);
}

// deg -> deg^{-1/2} (0 where deg==0), in place
__global__ void rsqrt_kernel(float* __restrict__ deg, int n) {
    int i = blockIdx.x * blockDim.x + threadIdx.x;
    if (i < n) {
        float d = deg[i];
        deg[i] = (d > 0.0f) ? rsqrtf(d) : 0.0f;
    }
}

// wgt[e] = exp(-dist^2) * dinv[row] * dinv[col]
__global__ void weight_kernel(const int* __restrict__ row, const int* __restrict__ col,
                              const float* __restrict__ dist, const float* __restrict__ dinv,
                              float* __restrict__ wgt, int E) {
    int e = blockIdx.x * blockDim.x + threadIdx.x;
    if (e < E) {
        float d = dist[e];
        wgt[e] = expf(-d * d) * dinv[row[e]] * dinv[col[e]];
    }
}

// side[row[e]][k] += wgt[e] * x[col[e]][k]; 64 consecutive lanes = one edge
// (coalesced gather of x[col], coalesced atomic scatter to side[row]; both L2-resident)
__global__ void scatter_kernel(const int* __restrict__ row, const int* __restrict__ col,
                               const float* __restrict__ wgt, const float* __restrict__ x,
                               float* __restrict__ side, int E) {
    long long gid = (long long)blockIdx.x * blockDim.x + threadIdx.x;
    int e = (int)(gid >> 6);
    int k = (int)(gid & 63);
    if (e < E) {
        float w = wgt[e];
        int r = row[e];
        int c = col[e];
        atomicAdd(&side[r * D + k], w * x[c * D + k]);
    }
}

// In-place epilogue on d_out: out = side @ W^T + b, via v_wmma_f32_16x16x4_f32.
// One wave owns a 16-row block; all A fragments (full K=64) are loaded into
// registers before any store, so in-place is hazard-free.
__global__ void gemm_wmma(float* __restrict__ out, const float* __restrict__ Wm,
                          const float* __restrict__ bias, int nRowBlocks) {
    int lane = threadIdx.x & 31;
    int wave = threadIdx.x >> 5;
    int rb = blockIdx.x * (blockDim.x >> 5) + wave;
    if (rb >= nRowBlocks) return;  // uniform per wave -> EXEC stays all-1s for WMMA

    const int lrow = lane & 15;
    const int koff = (lane >= 16) ? 2 : 0;   // A layout: hi half-wave holds K=2,3

    // Load all A fragments for this 16-row block (16 steps of K=4)
    const float* arow = out + (rb * 16 + lrow) * D;
    v2f a[16];
#pragma unroll
    for (int t = 0; t < 16; ++t) {
        a[t].x = arow[4 * t + koff];
        a[t].y = arow[4 * t + koff + 1];
    }

#pragma unroll
    for (int ct = 0; ct < 4; ++ct) {         // 4 column tiles of 16 outputs
        int cb = 16 * ct;
        const float* wr = Wm + (cb + lrow) * D;  // B = W^T: same lane/reg pattern as A
        v8f acc = {};
#pragma unroll
        for (int t = 0; t < 16; ++t) {
            v2f bf;
            bf.x = wr[4 * t + koff];
            bf.y = wr[4 * t + koff + 1];
            acc = __builtin_amdgcn_wmma_f32_16x16x4_f32(
                /*neg_a=*/false, a[t], /*neg_b=*/false, bf,
                /*c_mod=*/(short)0, acc, /*reuse_a=*/false, /*reuse_b=*/false);
        }
        float bv = bias[cb + lrow];
#pragma unroll
        for (int v = 0; v < 8; ++v) {
            int m = (lane >= 16) ? (v + 8) : v;  // D layout: hi half-wave = rows 8..15
            out[(rb * 16 + m) * D + cb + lrow] = acc[v] + bv;
        }
    }
}

extern "C" void kernel_launch(void* const* d_in, const int* in_sizes, int n_in,
                              void* d_out, int out_size, void* d_ws, size_t ws_size,
                              hipStream_t stream) {
    const float* x    = (const float*)d_in[0];
    const int*   ei   = (const int*)d_in[1];
    const float* dist = (const float*)d_in[2];
    const float* Wm   = (const float*)d_in[3];
    const float* bias = (const float*)d_in[4];
    float* out = (float*)d_out;

    const int N = in_sizes[0] / D;   // 100000
    const int E = in_sizes[2];       // 1600000
    const int* row = ei;
    const int* col = ei + E;

    float* deg = (float*)d_ws;       // N floats (becomes deg^{-1/2})
    float* wgt = deg + N;            // E floats

    const int T = 256;

    zero_f32<<<(N + T - 1) / T, T, 0, stream>>>(deg, N);
    zero_f32<<<(N * D + T - 1) / T, T, 0, stream>>>(out, N * D);

    degree_kernel<<<(E + T - 1) / T, T, 0, stream>>>(col, deg, E);
    rsqrt_kernel<<<(N + T - 1) / T, T, 0, stream>>>(deg, N);
    weight_kernel<<<(E + T - 1) / T, T, 0, stream>>>(row, col, dist, deg, wgt, E);

    long long total = (long long)E * D;   // 102.4M lanes, lane-per-(edge,feature)
    scatter_kernel<<<(int)((total + T - 1) / T), T, 0, stream>>>(row, col, wgt, x, out, E);

    const int nRB = N / 16;               // 6250 row blocks
    const int wpb = T / 32;               // 8 waves per block
    gemm_wmma<<<(nRB + wpb - 1) / wpb, T, 0, stream>>>(out, Wm, bias, nRB);
}